// SRLN_7945689497741
// MI455X (gfx1250) — compile-verified
//
#include <hip/hip_runtime.h>
#include <math.h>

// ---------------------------------------------------------------------------
// SPDNet forward + Karcher-mean center loss for 9x9 SPD matrices, B=65536.
// One sample per lane; all 9x9 matrices fully register-resident (unrolled
// constant indices). WMMA (v_wmma_f32_16x16x4_f32) used for the logits GEMM.
// ---------------------------------------------------------------------------

#define NSW 7           // Jacobi sweeps (quadratic convergence; ample for 9x9 fp32)
#define EPS_RE 1e-4f
#define NMAT 9
#define KARCHER_ITERS 6

typedef float v2f __attribute__((ext_vector_type(2)));
typedef float v8f __attribute__((ext_vector_type(8)));

// ---------------- per-lane 9x9 linear algebra (register resident) ----------

// Cyclic Jacobi eigensolver. A (81) destroyed; diagonal holds eigenvalues.
// V (81) receives eigenvectors as columns: A_in = V diag(lam) V^T.
__device__ __forceinline__ void eig9(float* A, float* V) {
#pragma unroll
  for (int i = 0; i < 9; ++i)
#pragma unroll
    for (int j = 0; j < 9; ++j) V[i * 9 + j] = (i == j) ? 1.f : 0.f;

#pragma unroll 1
  for (int sw = 0; sw < NSW; ++sw) {
#pragma unroll
    for (int p = 0; p < 8; ++p) {
#pragma unroll
      for (int q = p + 1; q < 9; ++q) {
        float apq = A[p * 9 + q];
        float app = A[p * 9 + p];
        float aqq = A[q * 9 + q];
        bool tiny = fabsf(apq) < 1e-26f;
        float denom = tiny ? 1.f : apq;
        float theta = (aqq - app) * 0.5f / denom;
        float t = copysignf(1.f, theta) / (fabsf(theta) + sqrtf(theta * theta + 1.f));
        t = tiny ? 0.f : t;
        float c = 1.f / sqrtf(t * t + 1.f);
        float s = t * c;
        // right-multiply by J (columns p,q)
#pragma unroll
        for (int k = 0; k < 9; ++k) {
          float akp = A[k * 9 + p], akq = A[k * 9 + q];
          A[k * 9 + p] = c * akp - s * akq;
          A[k * 9 + q] = s * akp + c * akq;
        }
        // left-multiply by J^T (rows p,q)
#pragma unroll
        for (int k = 0; k < 9; ++k) {
          float apk = A[p * 9 + k], aqk = A[q * 9 + k];
          A[p * 9 + k] = c * apk - s * aqk;
          A[q * 9 + k] = s * apk + c * aqk;
        }
        // V <- V J
#pragma unroll
        for (int k = 0; k < 9; ++k) {
          float vkp = V[k * 9 + p], vkq = V[k * 9 + q];
          V[k * 9 + p] = c * vkp - s * vkq;
          V[k * 9 + q] = s * vkp + c * vkq;
        }
      }
    }
  }
}

// B = V diag(f) V^T  (symmetric reconstruction)
__device__ __forceinline__ void recon9(const float* V, const float* f, float* B) {
#pragma unroll
  for (int i = 0; i < 9; ++i) {
    float w[9];
#pragma unroll
    for (int k = 0; k < 9; ++k) w[k] = V[i * 9 + k] * f[k];
#pragma unroll
    for (int j = i; j < 9; ++j) {
      float acc = 0.f;
#pragma unroll
      for (int k = 0; k < 9; ++k) acc = fmaf(w[k], V[j * 9 + k], acc);
      B[i * 9 + j] = acc;
      B[j * 9 + i] = acc;
    }
  }
}

// X <- W^T X W   (W in LDS / any memory, X in registers)
__device__ __forceinline__ void bimap9(float* X, const float* W) {
  float T[81];
#pragma unroll
  for (int i = 0; i < 9; ++i)
#pragma unroll
    for (int k = 0; k < 9; ++k) {
      float acc = 0.f;
#pragma unroll
      for (int j = 0; j < 9; ++j) acc = fmaf(W[j * 9 + i], X[j * 9 + k], acc);
      T[i * 9 + k] = acc;
    }
#pragma unroll
  for (int i = 0; i < 9; ++i)
#pragma unroll
    for (int l = 0; l < 9; ++l) {
      float acc = 0.f;
#pragma unroll
      for (int k = 0; k < 9; ++k) acc = fmaf(T[i * 9 + k], W[k * 9 + l], acc);
      X[i * 9 + l] = acc;
    }
}

// X <- C X C  with symmetric C (C^T = C)
__device__ __forceinline__ void sandwich9(float* X, const float* C) {
  float T[81];
#pragma unroll
  for (int i = 0; i < 9; ++i)
#pragma unroll
    for (int k = 0; k < 9; ++k) {
      float acc = 0.f;
#pragma unroll
      for (int j = 0; j < 9; ++j) acc = fmaf(C[i * 9 + j], X[j * 9 + k], acc);
      T[i * 9 + k] = acc;
    }
#pragma unroll
  for (int i = 0; i < 9; ++i)
#pragma unroll
    for (int l = 0; l < 9; ++l) {
      float acc = 0.f;
#pragma unroll
      for (int k = 0; k < 9; ++k) acc = fmaf(T[i * 9 + k], C[k * 9 + l], acc);
      X[i * 9 + l] = acc;
    }
}

// O = A * B (9x9)
__device__ __forceinline__ void mm9(const float* A, const float* B, float* O) {
#pragma unroll
  for (int i = 0; i < 9; ++i)
#pragma unroll
    for (int j = 0; j < 9; ++j) {
      float acc = 0.f;
#pragma unroll
      for (int k = 0; k < 9; ++k) acc = fmaf(A[i * 9 + k], B[k * 9 + j], acc);
      O[i * 9 + j] = acc;
    }
}

// SPD inverse via Cholesky: Ai = (L L^T)^-1 = L^-T L^-1
__device__ __forceinline__ void cholinv9(const float* A, float* Ai) {
  float L[81];
#pragma unroll
  for (int j = 0; j < 9; ++j) {
    float d = A[j * 9 + j];
#pragma unroll
    for (int k = 0; k < j; ++k) d = fmaf(-L[j * 9 + k], L[j * 9 + k], d);
    d = sqrtf(fmaxf(d, 1e-30f));
    L[j * 9 + j] = d;
    float invd = 1.f / d;
#pragma unroll
    for (int i = j + 1; i < 9; ++i) {
      float s = A[i * 9 + j];
#pragma unroll
      for (int k = 0; k < j; ++k) s = fmaf(-L[i * 9 + k], L[j * 9 + k], s);
      L[i * 9 + j] = s * invd;
    }
  }
  float Li[81];
#pragma unroll
  for (int j = 0; j < 9; ++j) {
    Li[j * 9 + j] = 1.f / L[j * 9 + j];
#pragma unroll
    for (int i = j + 1; i < 9; ++i) {
      float s = 0.f;
#pragma unroll
      for (int k = j; k < i; ++k) s = fmaf(L[i * 9 + k], Li[k * 9 + j], s);
      Li[i * 9 + j] = -s / L[i * 9 + i];
    }
  }
#pragma unroll
  for (int i = 0; i < 9; ++i)
#pragma unroll
    for (int j = i; j < 9; ++j) {
      float s = 0.f;
#pragma unroll
      for (int k = j; k < 9; ++k) s = fmaf(Li[k * 9 + i], Li[k * 9 + j], s);
      Ai[i * 9 + j] = s;
      Ai[j * 9 + i] = s;
    }
}

// ---------------- kernels ---------------------------------------------------

// bimap->reeig->bimap->reeig->bimap ; store h ; logm(h) -> feat
__global__ __launch_bounds__(256) void k_forward(
    const float* __restrict__ x, const float* __restrict__ W1,
    const float* __restrict__ W2, const float* __restrict__ W3,
    float* __restrict__ h_out, float* __restrict__ feat_out, int B) {
  __shared__ float sW1[81], sW2[81], sW3[81];
  int t = threadIdx.x;
  if (t < 81) { sW1[t] = W1[t]; sW2[t] = W2[t]; sW3[t] = W3[t]; }
  __syncthreads();
  int b = blockIdx.x * 256 + t;
  if (b >= B) return;

  float A[81], Vv[81], f[9];
  const float* xb = x + (size_t)b * 81;
#pragma unroll
  for (int i = 0; i < 81; ++i) A[i] = xb[i];

  bimap9(A, sW1);
  eig9(A, Vv);
#pragma unroll
  for (int k = 0; k < 9; ++k) f[k] = fmaxf(A[k * 9 + k], EPS_RE);
  recon9(Vv, f, A);

  bimap9(A, sW2);
  eig9(A, Vv);
#pragma unroll
  for (int k = 0; k < 9; ++k) f[k] = fmaxf(A[k * 9 + k], EPS_RE);
  recon9(Vv, f, A);

  bimap9(A, sW3);  // A = h
  float* hb = h_out + (size_t)b * 81;
#pragma unroll
  for (int i = 0; i < 81; ++i) hb[i] = A[i];

  eig9(A, Vv);
#pragma unroll
  for (int k = 0; k < 9; ++k) f[k] = logf(fmaxf(A[k * 9 + k], 1e-30f));
  recon9(Vv, f, A);  // A = logm(h)
  float* fb = feat_out + (size_t)b * 81;
#pragma unroll
  for (int i = 0; i < 81; ++i) fb[i] = A[i];
}

// logits = feat[B,81] @ Wl^T[81,2] + bl  via V_WMMA_F32_16X16X4_F32.
// One wave handles a 16-row tile; K padded 81->84 (21 WMMAs), N padded 2->16.
// B fragments come from a zero-padded LDS image of Wl^T: no divergent loads
// (and no exec-mask save/restore) inside the WMMA loop.
__global__ __launch_bounds__(256) void k_logits(
    const float* __restrict__ feat, const float* __restrict__ Wl,
    const float* __restrict__ bl, float* __restrict__ out, int B) {
  __shared__ float sB[84 * 16];  // sB[k][n] = (k<81 && n<2) ? Wl[n*81+k] : 0
  int t = threadIdx.x;
  for (int idx = t; idx < 84 * 16; idx += 256) {
    int k = idx >> 4, n = idx & 15;
    sB[idx] = (k < 81 && n < 2) ? Wl[n * 81 + k] : 0.f;
  }
  __syncthreads();  // before any exit: all threads reach the barrier

  int wave = t >> 5;
  int lane = t & 31;
  int tile = blockIdx.x * 8 + wave;
  int row0 = tile * 16;
  if (row0 >= B) return;  // wave-uniform: EXEC stays all-ones for WMMA

  int mrow = lane & 15;             // A-fragment row (ISA 32-bit A 16x4 layout)
  int kofs = (lane < 16) ? 0 : 2;   // lanes 16-31 hold K+2
  int n = lane & 15;                // B/D column
  int rrow = min(row0 + mrow, B - 1);
  const float* frow = feat + (size_t)rrow * 81;

  v8f acc;
#pragma unroll
  for (int r = 0; r < 8; ++r) acc[r] = 0.f;

#pragma unroll
  for (int k0 = 0; k0 < 84; k0 += 4) {
    int ka = k0 + kofs;
    v2f a, bm;
    a.x = (ka < 81) ? frow[ka] : 0.f;       // only the k0=80 tail is guarded
    a.y = (ka + 1 < 81) ? frow[ka + 1] : 0.f;
    bm.x = sB[ka * 16 + n];                 // unconditional LDS broadcast
    bm.y = sB[(ka + 1) * 16 + n];
    acc = __builtin_amdgcn_wmma_f32_16x16x4_f32(false, a, false, bm,
                                                (short)0, acc, false, false);
  }
  if (n < 2) {
    float bias = bl[n];
#pragma unroll
    for (int r = 0; r < 8; ++r) {
      int m = r + ((lane < 16) ? 0 : 8);  // D layout: VGPR r -> rows r / r+8
      int row = row0 + m;
      if (row < B) out[(size_t)row * 2 + n] = acc[r] + bias;
    }
  }
}

// Stage 1 of class-weighted sums: part[e][slice][c]. e==81 row accumulates the
// class counts (value 1.0). Deterministic (fixed-order shuffles + LDS tree).
__global__ __launch_bounds__(256) void k_class_sums(
    const float* __restrict__ h, const int* __restrict__ y,
    float* __restrict__ part, int B) {
  int e = blockIdx.x;      // 0..81
  int slice = blockIdx.y;  // 0..63
  int t = threadIdx.x;
  float a0 = 0.f, a1 = 0.f;
  for (int s = slice * 256 + t; s < B; s += 64 * 256) {
    float v = (e < 81) ? h[(size_t)s * 81 + e] : 1.f;
    int c = y[s];
    a0 += (c == 0) ? v : 0.f;
    a1 += (c == 1) ? v : 0.f;
  }
#pragma unroll
  for (int off = 16; off; off >>= 1) {
    a0 += __shfl_xor(a0, off);
    a1 += __shfl_xor(a1, off);
  }
  __shared__ float w0[8], w1[8];
  int lane = t & 31, wave = t >> 5;
  if (lane == 0) { w0[wave] = a0; w1[wave] = a1; }
  __syncthreads();
  if (t == 0) {
    float s0 = 0.f, s1 = 0.f;
#pragma unroll
    for (int w = 0; w < 8; ++w) { s0 += w0[w]; s1 += w1[w]; }
    part[(e * 64 + slice) * 2 + 0] = s0;
    part[(e * 64 + slice) * 2 + 1] = s1;
  }
}

// Stage 2: C0[c] = sum / max(count,1); zero Karcher accumulators and loss.
__global__ void k_finalize_mean(const float* __restrict__ part,
                                float* __restrict__ C, float* __restrict__ wsum,
                                float* __restrict__ acc,
                                float* __restrict__ loss_out) {
  int t = threadIdx.x;
  if (t < 2) {
    float s = 0.f;
    for (int sl = 0; sl < 64; ++sl) s += part[(81 * 64 + sl) * 2 + t];
    wsum[t] = fmaxf(s, 1.f);
  }
  __syncthreads();
  if (t < 162) {
    int c = t / 81, e = t % 81;
    float s = 0.f;
    for (int sl = 0; sl < 64; ++sl) s += part[(e * 64 + sl) * 2 + c];
    C[t] = s / wsum[c];
    acc[t] = 0.f;
  }
  if (t == 200) loss_out[0] = 0.f;
}

// Cis = C^-1/2, Cs = C^1/2 per class (one tiny eig per class).
__global__ void k_karcher_prep(const float* __restrict__ C,
                               float* __restrict__ Cis, float* __restrict__ Cs) {
  int c = threadIdx.x;
  if (c >= 2) return;
  float A[81], Vv[81], fs[9], fis[9];
#pragma unroll
  for (int i = 0; i < 81; ++i) A[i] = C[c * 81 + i];
  eig9(A, Vv);
#pragma unroll
  for (int k = 0; k < 9; ++k) {
    float l = fmaxf(A[k * 9 + k], 1e-30f);
    float sq = sqrtf(l);
    fs[k] = sq;
    fis[k] = 1.f / sq;
  }
  recon9(Vv, fis, A);
#pragma unroll
  for (int i = 0; i < 81; ++i) Cis[c * 81 + i] = A[i];
  recon9(Vv, fs, A);
#pragma unroll
  for (int i = 0; i < 81; ++i) Cs[c * 81 + i] = A[i];
}

// Per sample: Y = Cis[c] h Cis[c]; accumulate w * logm(Y) per class.
__global__ __launch_bounds__(256) void k_karcher_accum(
    const float* __restrict__ h, const int* __restrict__ y,
    const float* __restrict__ Cis, float* __restrict__ accum, int B) {
  __shared__ float sCis[162];
  __shared__ float sacc[162];
  int t = threadIdx.x;
  if (t < 162) { sCis[t] = Cis[t]; sacc[t] = 0.f; }
  __syncthreads();

  int b = blockIdx.x * 256 + t;
  bool valid = b < B;
  int c = 0;
  float A[81], Vv[81];
  if (valid) {
    c = y[b];
    const float* hb = h + (size_t)b * 81;
#pragma unroll
    for (int i = 0; i < 81; ++i) A[i] = hb[i];
    sandwich9(A, sCis + c * 81);
    eig9(A, Vv);
    float f[9];
#pragma unroll
    for (int k = 0; k < 9; ++k) f[k] = logf(fmaxf(A[k * 9 + k], 1e-30f));
    recon9(Vv, f, A);  // A = logm(Y)
  } else {
#pragma unroll
    for (int i = 0; i < 81; ++i) A[i] = 0.f;
  }

  int lane = t & 31;
#pragma unroll
  for (int e = 0; e < 81; ++e) {
    float v0 = (valid && c == 0) ? A[e] : 0.f;
    float v1 = (valid && c == 1) ? A[e] : 0.f;
#pragma unroll
    for (int off = 16; off; off >>= 1) {
      v0 += __shfl_xor(v0, off);
      v1 += __shfl_xor(v1, off);
    }
    if (lane == 0) {
      atomicAdd(&sacc[e], v0);
      atomicAdd(&sacc[81 + e], v1);
    }
  }
  __syncthreads();
  if (t < 162) atomicAdd(&accum[t], sacc[t]);
}

// C <- Cs expm(accum/wsum) Cs ; reset accumulators for next iteration.
__global__ void k_karcher_update(float* __restrict__ acc,
                                 const float* __restrict__ wsum,
                                 const float* __restrict__ Cs,
                                 float* __restrict__ C) {
  int t = threadIdx.x;
  if (t < 2) {
    int c = t;
    float inv = 1.f / wsum[c];
    float A[81], Vv[81], f[9], E[81], T[81];
#pragma unroll
    for (int i = 0; i < 81; ++i) A[i] = acc[c * 81 + i] * inv;
    eig9(A, Vv);
#pragma unroll
    for (int k = 0; k < 9; ++k) f[k] = expf(A[k * 9 + k]);
    recon9(Vv, f, E);
    mm9(Cs + c * 81, E, T);
    mm9(T, Cs + c * 81, A);
#pragma unroll
    for (int i = 0; i < 81; ++i) C[c * 81 + i] = A[i];
  }
  __syncthreads();
  if (t < 162) acc[t] = 0.f;
}

__global__ void k_minv(const float* __restrict__ C, float* __restrict__ Minv) {
  int c = threadIdx.x;
  if (c >= 2) return;
  float A[81], Ai[81];
#pragma unroll
  for (int i = 0; i < 81; ++i) A[i] = C[c * 81 + i];
  cholinv9(A, Ai);
#pragma unroll
  for (int i = 0; i < 81; ++i) Minv[c * 81 + i] = Ai[i];
}

// loss += sum_b ( 0.5*(tr(Minv[c] h_b) + tr(h_b^-1 M[c])) - 9 ) / B
__global__ __launch_bounds__(256) void k_loss(
    const float* __restrict__ h, const int* __restrict__ y,
    const float* __restrict__ M, const float* __restrict__ Minv,
    float* __restrict__ loss, float invB, int B) {
  __shared__ float sM[162], sMi[162];
  __shared__ float sl;
  int t = threadIdx.x;
  if (t < 162) { sM[t] = M[t]; sMi[t] = Minv[t]; }
  if (t == 0) sl = 0.f;
  __syncthreads();

  int b = blockIdx.x * 256 + t;
  float d = 0.f;
  if (b < B) {
    int c = y[b];
    float X[81], Xi[81];
    const float* hb = h + (size_t)b * 81;
#pragma unroll
    for (int i = 0; i < 81; ++i) X[i] = hb[i];
    cholinv9(X, Xi);
    float tr1 = 0.f, tr2 = 0.f;
#pragma unroll
    for (int i = 0; i < 81; ++i) {
      tr1 = fmaf(sMi[c * 81 + i], X[i], tr1);   // tr(Minv X), both symmetric
      tr2 = fmaf(Xi[i], sM[c * 81 + i], tr2);   // tr(Xinv M)
    }
    d = (0.5f * (tr1 + tr2) - (float)NMAT) * invB;
  }
#pragma unroll
  for (int off = 16; off; off >>= 1) d += __shfl_xor(d, off);
  if ((t & 31) == 0) atomicAdd(&sl, d);
  __syncthreads();
  if (t == 0) atomicAdd(loss, sl);
}

// ---------------- launcher ---------------------------------------------------

extern "C" void kernel_launch(void* const* d_in, const int* in_sizes, int n_in,
                              void* d_out, int out_size, void* d_ws,
                              size_t ws_size, hipStream_t stream) {
  (void)n_in; (void)out_size; (void)ws_size;
  const float* x  = (const float*)d_in[0];
  const int*   y  = (const int*)d_in[1];
  const float* W1 = (const float*)d_in[2];
  const float* W2 = (const float*)d_in[3];
  const float* W3 = (const float*)d_in[4];
  const float* Wl = (const float*)d_in[5];
  const float* bl = (const float*)d_in[6];
  int B = in_sizes[0] / 81;

  float* out  = (float*)d_out;
  float* loss = out + (size_t)B * 2;  // tuple: (logits[B,2], cl)

  float* ws   = (float*)d_ws;
  float* h    = ws;                        // B*81
  float* feat = h + (size_t)B * 81;        // B*81
  float* part = feat + (size_t)B * 81;     // 82*64*2
  float* C    = part + 82 * 64 * 2;        // 2*81
  float* Cis  = C + 162;                   // 2*81
  float* Cs   = Cis + 162;                 // 2*81
  float* Minv = Cs + 162;                  // 2*81
  float* wsum = Minv + 162;                // 2
  float* acc  = wsum + 2;                  // 2*81

  int nblk = (B + 255) / 256;

  k_forward<<<nblk, 256, 0, stream>>>(x, W1, W2, W3, h, feat, B);
  k_logits<<<(B + 127) / 128, 256, 0, stream>>>(feat, Wl, bl, out, B);

  dim3 g2(82, 64);
  k_class_sums<<<g2, 256, 0, stream>>>(h, y, part, B);
  k_finalize_mean<<<1, 256, 0, stream>>>(part, C, wsum, acc, loss);

  for (int it = 0; it < KARCHER_ITERS; ++it) {
    k_karcher_prep<<<1, 64, 0, stream>>>(C, Cis, Cs);
    k_karcher_accum<<<nblk, 256, 0, stream>>>(h, y, Cis, acc, B);
    k_karcher_update<<<1, 192, 0, stream>>>(acc, wsum, Cs, C);
  }

  k_minv<<<1, 64, 0, stream>>>(C, Minv);
  k_loss<<<nblk, 256, 0, stream>>>(h, y, C, Minv, loss, 1.0f / (float)B, B);
}